// HOGCN_32925219291717
// MI455X (gfx1250) — compile-verified
//
#include <hip/hip_runtime.h>
#include <hip/hip_bf16.h>

typedef float v2f __attribute__((ext_vector_type(2)));
typedef float v8f __attribute__((ext_vector_type(8)));

#define D_FEAT 64

// ---------------------------------------------------------------------------
// Zero a float buffer (float4-vectorized). Needed because d_ws is poisoned.
// ---------------------------------------------------------------------------
__global__ __launch_bounds__(256) void gcn_zero(float4* __restrict__ p, long long n4) {
    long long i = (long long)blockIdx.x * blockDim.x + threadIdx.x;
    if (i < n4) p[i] = make_float4(0.f, 0.f, 0.f, 0.f);
}

// ---------------------------------------------------------------------------
// Edge scatter: agg[dst] += w * x[src].  One thread per (edge, 4-feature
// chunk): float4 gather (L2-resident, x is only 12.8MB) + 4 non-returning
// global_atomic_add_f32.
// ---------------------------------------------------------------------------
__global__ __launch_bounds__(256) void gcn_scatter(const long long* __restrict__ src,
                                                   const long long* __restrict__ dst,
                                                   const float* __restrict__ ew,
                                                   const float* __restrict__ xin,
                                                   float* __restrict__ agg,
                                                   long long E) {
    long long tid = (long long)blockIdx.x * blockDim.x + threadIdx.x;
    if (tid >= E * 16) return;
    long long e = tid >> 4;
    int c = (int)(tid & 15) << 2;           // feature chunk base (0..60)

    long long s = src[e];
    long long d = dst[e];
    float w = ew[e];

    const float4 v = *(const float4*)(xin + s * D_FEAT + c);
    float* ap = agg + d * D_FEAT + c;
    atomicAdd(ap + 0, w * v.x);
    atomicAdd(ap + 1, w * v.y);
    atomicAdd(ap + 2, w * v.z);
    atomicAdd(ap + 3, w * v.w);
}

// ---------------------------------------------------------------------------
// Fused dense phase, fp32 WMMA:
//   out = relu( agg @ Wrel^T + brel + xin @ Wroot^T )
// Block = 4 waves; blockIdx.x picks a 16-row slab, wave picks the 16-col tile.
// K-loop: 16 steps of V_WMMA_F32_16X16X4_F32 per weight matrix (32 chained
// WMMAs into one v8f accumulator).
//
// fp32 WMMA lane layout (ISA 7.12.2):
//   A 16x4 : lane<16 -> {K=0,K=1}, lane>=16 -> {K=2,K=3}, row M = lane&15
//   B 4x16 : same pattern with column N = lane&15
//   => both are contiguous float2 loads at (row * 64 + k0 + 2*(lane>>4))
//   C/D    : VGPR r, lane<16 -> (M=r, N=lane); lane>=16 -> (M=r+8, N=lane-16)
// ---------------------------------------------------------------------------
__global__ __launch_bounds__(128) void gcn_gemm_wmma(const float* __restrict__ agg,
                                                     const float* __restrict__ xin,
                                                     const float* __restrict__ Wrel,
                                                     const float* __restrict__ Wroot,
                                                     const float* __restrict__ brel,
                                                     float* __restrict__ out,
                                                     int N) {
    const int wave  = threadIdx.x >> 5;      // 0..3 -> column tile
    const int lane  = threadIdx.x & 31;
    const int lrow  = lane & 15;
    const int khalf = lane >> 4;             // 0 or 1

    const int row0 = blockIdx.x * 16;        // N is an exact multiple of 16
    const int col0 = wave * 16;
    if (row0 >= N) return;

    v8f c = {0.f, 0.f, 0.f, 0.f, 0.f, 0.f, 0.f, 0.f};

    const float* aRow  = agg   + (size_t)(row0 + lrow) * D_FEAT + 2 * khalf;
    const float* xRow  = xin   + (size_t)(row0 + lrow) * D_FEAT + 2 * khalf;
    const float* bRel  = Wrel  + (size_t)(col0 + lrow) * D_FEAT + 2 * khalf;
    const float* bRoot = Wroot + (size_t)(col0 + lrow) * D_FEAT + 2 * khalf;

#pragma unroll
    for (int k0 = 0; k0 < D_FEAT; k0 += 4) {
        v2f a0 = *(const v2f*)(aRow + k0);
        v2f b0 = *(const v2f*)(bRel + k0);
        c = __builtin_amdgcn_wmma_f32_16x16x4_f32(false, a0, false, b0,
                                                  (short)0, c, false, false);
        v2f a1 = *(const v2f*)(xRow + k0);
        v2f b1 = *(const v2f*)(bRoot + k0);
        c = __builtin_amdgcn_wmma_f32_16x16x4_f32(false, a1, false, b1,
                                                  (short)0, c, false, false);
    }

    const float bias = brel[col0 + lrow];    // bias indexed by output column N
    float* oBase = out + (size_t)row0 * D_FEAT + col0 + lrow;
#pragma unroll
    for (int r = 0; r < 8; ++r) {
        float v = c[r] + bias;
        v = v > 0.f ? v : 0.f;
        oBase[(size_t)(r + 8 * khalf) * D_FEAT] = v;   // relu + store
    }
}

// ---------------------------------------------------------------------------
// Launcher.  Inputs (setup_inputs order):
//   0:x [N,64] f32   1:edge_index [2,E] i64   2:edge_weight [E] f32
//   3:W_rel1 [64,64] 4:b_rel1 [64] 5:W_root1 [64,64]
//   6:W_rel2 [64,64] 7:b_rel2 [64] 8:W_root2 [64,64]
// Workspace: agg (N*64 f32) | h1 (N*64 f32)  -> 25.6 MB
// ---------------------------------------------------------------------------
extern "C" void kernel_launch(void* const* d_in, const int* in_sizes, int n_in,
                              void* d_out, int out_size, void* d_ws, size_t ws_size,
                              hipStream_t stream) {
    const float*     x   = (const float*)d_in[0];
    const long long* ei  = (const long long*)d_in[1];
    const float*     ew  = (const float*)d_in[2];
    const float*     Wr1 = (const float*)d_in[3];
    const float*     br1 = (const float*)d_in[4];
    const float*     Wo1 = (const float*)d_in[5];
    const float*     Wr2 = (const float*)d_in[6];
    const float*     br2 = (const float*)d_in[7];
    const float*     Wo2 = (const float*)d_in[8];
    float*           out = (float*)d_out;

    const int       N = in_sizes[0] / D_FEAT;
    const long long E = in_sizes[2];

    const long long* src = ei;        // edge_index[0]
    const long long* dst = ei + E;    // edge_index[1]

    float* agg = (float*)d_ws;                       // N*64 floats
    float* h1  = (float*)d_ws + (size_t)N * D_FEAT;  // N*64 floats

    const long long nd  = (long long)N * D_FEAT;
    const long long nd4 = nd / 4;
    dim3 zgrid((unsigned)((nd4 + 255) / 256));
    dim3 sgrid((unsigned)((E * 16 + 255) / 256));
    dim3 ggrid((unsigned)(N / 16));                  // 50000/16 = 3125 exact

    // ---- Layer 1 ----
    gcn_zero<<<zgrid, 256, 0, stream>>>((float4*)agg, nd4);
    gcn_scatter<<<sgrid, 256, 0, stream>>>(src, dst, ew, x, agg, E);
    gcn_gemm_wmma<<<ggrid, 128, 0, stream>>>(agg, x, Wr1, Wo1, br1, h1, N);

    // ---- Layer 2 ----
    gcn_zero<<<zgrid, 256, 0, stream>>>((float4*)agg, nd4);
    gcn_scatter<<<sgrid, 256, 0, stream>>>(src, dst, ew, h1, agg, E);
    gcn_gemm_wmma<<<ggrid, 128, 0, stream>>>(agg, h1, Wr2, Wo2, br2, out, N);
}